// GCN_3753801416995
// MI455X (gfx1250) — compile-verified
//
#include <hip/hip_runtime.h>

typedef __attribute__((ext_vector_type(2))) float v2f;
typedef __attribute__((ext_vector_type(8))) float v8f;

#define D_IN  128
#define D_H   128
#define D_OUT 64
#define BN_EPS 1e-5f

// ---------------------------------------------------------------------------
// GEMM: C[nrows x KOUT] = A[nrows x 128] @ W[128 x KOUT] using
// V_WMMA_F32_16X16X4_F32 (exact f32, matches reference precision).
// Block = 256 threads = 8 waves; each wave computes a 16-row strip across all
// KOUT columns (KOUT/16 accumulators). W lives in LDS packed as K-pairs so a
// lane's B fragment is one ds_load_b64; A fragment is one global_load_b64
// reused across all N-tiles.
// ---------------------------------------------------------------------------
template <int KOUT>
__global__ __launch_bounds__(256) void gemm_wmma_f32(
    const float* __restrict__ A, const float* __restrict__ W,
    float* __restrict__ C, int nrows)
{
    // Wp[kp*KOUT + c] = { W[2kp][c], W[2kp+1][c] }   (kp = 0..63)
    __shared__ v2f Wp[64 * KOUT];
    const int tid = threadIdx.x;
    for (int i = tid; i < 64 * KOUT; i += 256) {
        int kp = i / KOUT, c = i - kp * KOUT;
        v2f w;
        w.x = W[(2 * kp + 0) * KOUT + c];
        w.y = W[(2 * kp + 1) * KOUT + c];
        Wp[i] = w;
    }
    __syncthreads();

    const int lane  = tid & 31;
    const int wv    = tid >> 5;
    const int mbase = (blockIdx.x * 8 + wv) * 16;
    if (mbase >= nrows) return;              // whole wave exits together

    const int mrow  = lane & 15;
    const int khalf = lane >> 4;             // 0 -> K{0,1}, 1 -> K{2,3}
    int arow = mbase + mrow;
    if (arow >= nrows) arow = nrows - 1;     // clamp loads; stores are guarded
    const float* __restrict__ Ar = A + (size_t)arow * 128;

    v8f zero;
#pragma unroll
    for (int j = 0; j < 8; ++j) zero[j] = 0.0f;
    v8f acc[KOUT / 16];
#pragma unroll
    for (int n = 0; n < KOUT / 16; ++n) acc[n] = zero;

    for (int kk = 0; kk < 32; ++kk) {        // K = 128 in steps of 4
        v2f a = *(const v2f*)(Ar + 4 * kk + 2 * khalf);
        const v2f* __restrict__ wrow = &Wp[(2 * kk + khalf) * KOUT + (lane & 15)];
#pragma unroll
        for (int n = 0; n < KOUT / 16; ++n) {
            v2f b = wrow[n * 16];
            acc[n] = __builtin_amdgcn_wmma_f32_16x16x4_f32(
                false, a, false, b, (short)0, acc[n], false, false);
        }
    }

    // C/D layout: VGPR r -> M = r (lanes 0-15), M = r+8 (lanes 16-31)
    const int rbase = mbase + 8 * khalf;
#pragma unroll
    for (int n = 0; n < KOUT / 16; ++n) {
        int col = n * 16 + (lane & 15);
#pragma unroll
        for (int r = 0; r < 8; ++r) {
            int row = rbase + r;
            if (row < nrows) C[(size_t)row * KOUT + col] = acc[n][r];
        }
    }
}

// ---------------------------------------------------------------------------
// Small helper kernels
// ---------------------------------------------------------------------------
__global__ void k_fill(float* p, float v, int n) {
    int i = blockIdx.x * blockDim.x + threadIdx.x;
    if (i < n) p[i] = v;
}

__global__ void k_deg(const long long* __restrict__ dst, float* deg, int E) {
    int e = blockIdx.x * blockDim.x + threadIdx.x;
    if (e < E) atomicAdd(&deg[(int)dst[e]], 1.0f);
}

__global__ void k_rsqrt(float* p, int n) {
    int i = blockIdx.x * blockDim.x + threadIdx.x;
    if (i < n) p[i] = rsqrtf(p[i]);
}

// out[i,f] = hw[i,f] * dis[i]^2 + bias[f]   (self-loop term; also fully
// initializes the output so the later edge atomics accumulate on top)
__global__ void k_agg_init(const float* __restrict__ hw,
                           const float* __restrict__ bias,
                           const float* __restrict__ dis,
                           float* __restrict__ out, int n, int D) {
    int i = blockIdx.x * blockDim.x + threadIdx.x;
    if (i < n * D) {
        int node = i / D, f = i - node * D;
        float d = dis[node];
        out[i] = hw[i] * d * d + bias[f];
    }
}

// One wave per edge: float4 gather of hw[src], scaled atomic scatter to out[dst]
__global__ __launch_bounds__(256) void k_agg_edges(
    const float* __restrict__ hw, const float* __restrict__ dis,
    const long long* __restrict__ src, const long long* __restrict__ dst,
    float* __restrict__ out, int E, int D)
{
    int gwave = (blockIdx.x * 256 + threadIdx.x) >> 5;
    int lane  = threadIdx.x & 31;
    if (gwave >= E) return;
    int s = (int)src[gwave];
    int d = (int)dst[gwave];
    float w = dis[s] * dis[d];
    const float4* __restrict__ hs = (const float4*)(hw + (size_t)s * D);
    float* od = out + (size_t)d * D;
    for (int q = lane; q < D / 4; q += 32) {
        float4 v = hs[q];
        atomicAdd(od + 4 * q + 0, v.x * w);
        atomicAdd(od + 4 * q + 1, v.y * w);
        atomicAdd(od + 4 * q + 2, v.z * w);
        atomicAdd(od + 4 * q + 3, v.w * w);
    }
}

// Per-feature sum / sumsq partials, atomically folded into stats[0..255]
__global__ __launch_bounds__(128) void k_bn_stats(
    const float* __restrict__ h, float* __restrict__ stats,
    int n, int rows_per_block)
{
    int f  = threadIdx.x;                         // 0..127
    int r0 = blockIdx.x * rows_per_block;
    int r1 = r0 + rows_per_block; if (r1 > n) r1 = n;
    float s = 0.0f, s2 = 0.0f;
    for (int r = r0; r < r1; ++r) {
        float v = h[(size_t)r * 128 + f];
        s += v; s2 += v * v;
    }
    atomicAdd(&stats[f], s);
    atomicAdd(&stats[128 + f], s2);
}

__global__ void k_bn_relu(float* __restrict__ h,
                          const float* __restrict__ stats,
                          const float* __restrict__ g,
                          const float* __restrict__ be, int n) {
    int i = blockIdx.x * blockDim.x + threadIdx.x;
    if (i < n * 128) {
        int f = i & 127;
        float inv_n = 1.0f / (float)n;
        float mu  = stats[f] * inv_n;
        float var = stats[128 + f] * inv_n - mu * mu;
        float v = (h[i] - mu) * rsqrtf(var + BN_EPS) * g[f] + be[f];
        h[i] = fmaxf(v, 0.0f);
    }
}

// ---------------------------------------------------------------------------
extern "C" void kernel_launch(void* const* d_in, const int* in_sizes, int n_in,
                              void* d_out, int out_size, void* d_ws, size_t ws_size,
                              hipStream_t stream)
{
    const float*     x   = (const float*)d_in[0];
    const long long* ei  = (const long long*)d_in[1];   // jnp.int64
    const float*     W1  = (const float*)d_in[2];
    const float*     b1  = (const float*)d_in[3];
    const float*     g1  = (const float*)d_in[4];
    const float*     be1 = (const float*)d_in[5];
    const float*     W2  = (const float*)d_in[6];
    const float*     b2  = (const float*)d_in[7];
    const float*     g2  = (const float*)d_in[8];
    const float*     be2 = (const float*)d_in[9];
    const float*     W3  = (const float*)d_in[10];
    const float*     b3  = (const float*)d_in[11];
    float*           out = (float*)d_out;

    const int N = in_sizes[0] / D_IN;
    const int E = in_sizes[1] / 2;
    const long long* src = ei;
    const long long* dst = ei + E;

    // workspace layout
    char* ws = (char*)d_ws;
    float* dis = (float*)ws;
    size_t off = ((size_t)N * 4 + 255) & ~(size_t)255;
    float* stats = (float*)(ws + off); off += 1024;          // 256 floats
    float* buf0  = (float*)(ws + off); off += (size_t)N * D_H * 4;
    float* buf1  = (float*)(ws + off);

    const dim3 T256(256);
    const int gemm_blocks = (N + 127) / 128;                 // 8 waves x 16 rows
    const int ndh_blocks  = (N * D_H + 255) / 256;
    const int edge_blocks = (E + 7) / 8;                     // 8 edges/block
    const int stat_blocks = (N + 255) / 256;

    // degrees: deg = 1 (self-loop) + in-degree; dis = deg^{-1/2}
    k_fill<<<(N + 255) / 256, T256, 0, stream>>>(dis, 1.0f, N);
    k_deg<<<(E + 255) / 256, T256, 0, stream>>>(dst, dis, E);
    k_rsqrt<<<(N + 255) / 256, T256, 0, stream>>>(dis, N);

    // ---- Layer 1: x @ W1 -> buf0; aggregate -> buf1; BN+ReLU in place ----
    gemm_wmma_f32<D_H><<<gemm_blocks, T256, 0, stream>>>(x, W1, buf0, N);
    k_agg_init<<<ndh_blocks, T256, 0, stream>>>(buf0, b1, dis, buf1, N, D_H);
    k_agg_edges<<<edge_blocks, T256, 0, stream>>>(buf0, dis, src, dst, buf1, E, D_H);
    k_fill<<<1, T256, 0, stream>>>(stats, 0.0f, 256);
    k_bn_stats<<<stat_blocks, dim3(128), 0, stream>>>(buf1, stats, N, 256);
    k_bn_relu<<<ndh_blocks, T256, 0, stream>>>(buf1, stats, g1, be1, N);

    // ---- Layer 2 ----
    gemm_wmma_f32<D_H><<<gemm_blocks, T256, 0, stream>>>(buf1, W2, buf0, N);
    k_agg_init<<<ndh_blocks, T256, 0, stream>>>(buf0, b2, dis, buf1, N, D_H);
    k_agg_edges<<<edge_blocks, T256, 0, stream>>>(buf0, dis, src, dst, buf1, E, D_H);
    k_fill<<<1, T256, 0, stream>>>(stats, 0.0f, 256);
    k_bn_stats<<<stat_blocks, dim3(128), 0, stream>>>(buf1, stats, N, 256);
    k_bn_relu<<<ndh_blocks, T256, 0, stream>>>(buf1, stats, g2, be2, N);

    // ---- Layer 3: h @ W3 -> buf0 (N x 64); aggregate -> d_out ----
    gemm_wmma_f32<D_OUT><<<gemm_blocks, T256, 0, stream>>>(buf1, W3, buf0, N);
    const int ndo_blocks = (N * D_OUT + 255) / 256;
    k_agg_init<<<ndo_blocks, T256, 0, stream>>>(buf0, b3, dis, out, N, D_OUT);
    k_agg_edges<<<edge_blocks, T256, 0, stream>>>(buf0, dis, src, dst, out, E, D_OUT);
}